// SinePredictor_60052232732736
// MI455X (gfx1250) — compile-verified
//
#include <hip/hip_runtime.h>
#include <hip/hip_bf16.h>

// SinePredictor edge scorer for MI455X (gfx1250, wave32).
//
//   score[e] = dot(sin(h[src[e]] - h[dst[e]]), W) + b      (D = 128)
//   out[0:E]   = pairwise softmax over (e&~1, e|1)
//   out[E:2E]  = (prob > 0.5) as float
//
// Reduction over D is done with a chain of 32 V_WMMA_F32_16X16X4_F32:
//   A(16x4)  = W broadcast along M  -> A[m][k] = W[4c+k]
//   B(4x16)  = sin(s-o) for 16 edges along N (one edge per column)
//   D[m][n]  = score[edge n] in every element -> lane reads acc[0], no shuffles.
//
// Per-lane fragments (wave32):
//   base = 4c + (lane<16 ? 0 : 2)
//   A frag = { W[base], W[base+1] }                 (LDS broadcast)
//   B frag = { sin(s-o)[base], sin(s-o)[base+1] }   (two float2 global gathers)
// (B row-striping assumed VGPR v: lanes0-15 -> K=v, lanes16-31 -> K=v+2,
//  mirroring the documented A/C/D layouts in CDNA5 ISA.)

typedef __attribute__((ext_vector_type(2))) float v2f;
typedef __attribute__((ext_vector_type(8))) float v8f;

#define DFEAT 128

__global__ __launch_bounds__(256) void sine_edge_score_kernel(
    const float* __restrict__ h,
    const long long* __restrict__ edges,   // [2, E] int64
    const float* __restrict__ W,           // [1, 128]
    const float* __restrict__ bias,        // [1]
    float* __restrict__ out,               // [2*E]: probs then mask-as-float
    int nEdges)
{
    __shared__ float ldsW[DFEAT];
    const int tid = threadIdx.x;
    if (tid < DFEAT) ldsW[tid] = W[tid];
    __syncthreads();

    const int lane = tid & 31;
    const int tile = blockIdx.x * (blockDim.x >> 5) + (tid >> 5);
    const int numTiles = nEdges >> 4;          // 16 edges per wave-tile
    if (tile >= numTiles) return;              // wave-uniform: EXEC stays full

    const int m    = lane & 15;                // edge slot (B-matrix column N)
    const int e    = tile * 16 + m;
    const int koff = (lane & 16) ? 2 : 0;      // which half of the K=4 chunk

    const long long src = edges[e];
    const long long dst = edges[e + nEdges];
    const float* __restrict__ rs = h + (size_t)src * DFEAT;
    const float* __restrict__ ro = h + (size_t)dst * DFEAT;

    v8f acc = {0.f, 0.f, 0.f, 0.f, 0.f, 0.f, 0.f, 0.f};

#pragma unroll
    for (int c = 0; c < DFEAT / 4; ++c) {
        const int base = c * 4 + koff;
        // gather: lanes L and L+16 read adjacent 8B of the same (random) row
        v2f s2 = *(const v2f*)(rs + base);
        v2f o2 = *(const v2f*)(ro + base);
        v2f bf;
        bf.x = __sinf(s2.x - o2.x);
        bf.y = __sinf(s2.y - o2.y);
        v2f af = *(const v2f*)(ldsW + base);   // LDS broadcast read
        // 8 args: (neg_a, A, neg_b, B, c_mod, C, reuse_a, reuse_b)
        acc = __builtin_amdgcn_wmma_f32_16x16x4_f32(
            false, af, false, bf, (short)0, acc, false, false);
    }

    // every acc element of this lane equals score[edge m]
    float s = acc[0] + bias[0];

    // pairwise softmax: partner edge (m^1) lives at lane (L^1)
    float p  = __shfl_xor(s, 1, 32);
    float mx = fmaxf(s, p);
    float es = __expf(s - mx);
    float ep = __expf(p - mx);
    float prob = es / (es + ep);

    if (lane < 16) {                           // lanes 16..31 hold duplicates
        out[e]          = prob;
        out[nEdges + e] = (prob > 0.5f) ? 1.0f : 0.0f;
    }
}

extern "C" void kernel_launch(void* const* d_in, const int* in_sizes, int n_in,
                              void* d_out, int out_size, void* d_ws, size_t ws_size,
                              hipStream_t stream) {
    const float*     h     = (const float*)d_in[0];
    const long long* edges = (const long long*)d_in[1];
    const float*     W     = (const float*)d_in[2];
    const float*     b     = (const float*)d_in[3];
    float*           out   = (float*)d_out;

    const int nEdges   = in_sizes[1] / 2;      // edges is [2, E] flat
    const int numTiles = nEdges / 16;          // one wave per 16-edge tile
    const int wavesPerBlock = 256 / 32;
    const int blocks = (numTiles + wavesPerBlock - 1) / wavesPerBlock;

    sine_edge_score_kernel<<<blocks, 256, 0, stream>>>(h, edges, W, b, out, nEdges);
}